// ContextKnowledgeEncoder_44195213475942
// MI455X (gfx1250) — compile-verified
//
#include <hip/hip_runtime.h>
#include <hip/hip_bf16.h>

// Problem constants (from reference)
constexpr int N  = 16;
constexpr int TS = 1024;
constexpr int K  = 32;
constexpr int TK = 128;
constexpr int D  = 512;
constexpr int TF = TK + TS;   // full sequence length = 1152

typedef __attribute__((ext_vector_type(2))) float v2f;
typedef __attribute__((ext_vector_type(8))) float v8f;

// ---------------------------------------------------------------------------
// Kernel 1: ctx_use[n,d] = sum_t mask * embed[src_tokens[n,t], d] / count
// grid (N, 4), block 128 -> each thread owns one of 512 dims.
// Prefix masks => branch is uniform per block; token/mask loads are scalar.
// ---------------------------------------------------------------------------
__global__ __launch_bounds__(128) void ctx_pool_kernel(
    const float* __restrict__ embed, const int* __restrict__ src_tokens,
    const int* __restrict__ src_mask, float* __restrict__ ctx_use) {
  const int n = blockIdx.x;
  const int d = blockIdx.y * 128 + threadIdx.x;
  const int* toks = src_tokens + (size_t)n * TS;
  const int* msks = src_mask + (size_t)n * TS;
  float acc = 0.0f;
  int cnt = 0;
  for (int t = 0; t < TS; ++t) {
    if (msks[t]) {
      acc += embed[(size_t)toks[t] * D + d];
      ++cnt;
    }
  }
  ctx_use[(size_t)n * D + d] = acc / (float)cnt;   // src_len >= TS/4 > 0
}

// ---------------------------------------------------------------------------
// Kernel 2: kn_use[n,k,d] = masked mean over TK tokens.
// One block per (n,k); 128 threads x float4 = full 512-dim row.
// Prefetch next gathered row (global_prefetch_b8) to hide gather latency.
// ---------------------------------------------------------------------------
__global__ __launch_bounds__(128) void kn_pool_kernel(
    const float* __restrict__ embed, const int* __restrict__ know_tokens,
    const int* __restrict__ know_mask, float* __restrict__ kn_use) {
  const int nk = blockIdx.x;                 // 0 .. N*K-1
  const int d0 = threadIdx.x * 4;
  const int* toks = know_tokens + (size_t)nk * TK;
  const int* msks = know_mask + (size_t)nk * TK;
  float4 acc = make_float4(0.f, 0.f, 0.f, 0.f);
  int cnt = 0;
  for (int t = 0; t < TK; ++t) {
    if (t + 1 < TK)
      __builtin_prefetch(embed + (size_t)toks[t + 1] * D + d0, 0, 1);
    if (msks[t]) {
      const float4 e = *(const float4*)(embed + (size_t)toks[t] * D + d0);
      acc.x += e.x; acc.y += e.y; acc.z += e.z; acc.w += e.w;
      ++cnt;
    }
  }
  const float inv = 1.0f / (float)cnt;       // know_len >= TK/4 > 0
  float4 r = make_float4(acc.x * inv, acc.y * inv, acc.z * inv, acc.w * inv);
  *(float4*)(kn_use + (size_t)nk * D + d0) = r;
}

// ---------------------------------------------------------------------------
// Kernel 3: ck_attn[n,k] = dot(kn_use[n,k,:], ctx_use[n,:]) via WMMA f32.
// One wave per n. A = 16 rows x 4 K-dims of kn_use (16x4 f32 A layout:
// lanes 0-15 -> K=0,1; lanes 16-31 -> K=2,3). B = ctx_use chunk broadcast
// across all 16 columns (column-identical, so B layout is safe). D/4 = 128
// accumulation steps in f32; two accumulators cover rows 0-15 and 16-31.
// C/D layout: lane l holds column N=l&15, rows M = r + 8*(l>>4).
// ---------------------------------------------------------------------------
__global__ __launch_bounds__(32) void ck_attn_wmma_kernel(
    const float* __restrict__ kn_use, const float* __restrict__ ctx_use,
    float* __restrict__ ck_attn) {
  const int n = blockIdx.x;
  const int lane = threadIdx.x;              // 0..31 (wave32)
  const int m = lane & 15;
  const int khalf = lane >> 4;               // 0 or 1
  const float* A0p = kn_use + ((size_t)n * K + m) * D;
  const float* A1p = kn_use + ((size_t)n * K + m + 16) * D;
  const float* Bp  = ctx_use + (size_t)n * D;
  v8f acc0 = {};
  v8f acc1 = {};
  for (int c = 0; c < D; c += 4) {
    const int o = c + 2 * khalf;             // even -> 8B-aligned v2f loads
    v2f a0 = *(const v2f*)(A0p + o);
    v2f a1 = *(const v2f*)(A1p + o);
    v2f b  = *(const v2f*)(Bp + o);
    acc0 = __builtin_amdgcn_wmma_f32_16x16x4_f32(
        false, a0, false, b, (short)0, acc0, false, false);
    acc1 = __builtin_amdgcn_wmma_f32_16x16x4_f32(
        false, a1, false, b, (short)0, acc1, false, false);
  }
  if (m == 0) {                              // lanes 0 and 16 hold column 0
    const int rbase = khalf * 8;
    float* out = ck_attn + (size_t)n * K;
#pragma unroll
    for (int r = 0; r < 8; ++r) {
      out[rbase + r]      = acc0[r];         // rows 0..15
      out[16 + rbase + r] = acc1[r];         // rows 16..31
    }
  }
}

// ---------------------------------------------------------------------------
// Kernel 4: per-example selected knowledge id.
// sel = use_cs_ids ? cs_ids[n] : argmax_k ck_attn[n,k]  (tie -> lowest k)
// One wave per n; K == warpSize == 32 -> shuffle argmax.
// ---------------------------------------------------------------------------
__global__ __launch_bounds__(32) void select_id_kernel(
    const float* __restrict__ ck_attn, const int* __restrict__ cs_ids,
    const int* __restrict__ use_cs_ids, int* __restrict__ sel_ids) {
  const int n = blockIdx.x;
  const int lane = threadIdx.x;
  float v = ck_attn[(size_t)n * K + lane];
  int idx = lane;
  for (int off = 16; off > 0; off >>= 1) {
    float ov = __shfl_xor(v, off, 32);
    int oi = __shfl_xor(idx, off, 32);
    if (ov > v || (ov == v && oi < idx)) { v = ov; idx = oi; }
  }
  if (lane == 0)
    sel_ids[n] = (use_cs_ids[0] != 0) ? cs_ids[n] : idx;
}

// ---------------------------------------------------------------------------
// Kernel 5: full_enc = concat(cs_enc, ctx) with masked-zero rows;
// full_mask written as 0/1 floats. One block per output row; 128 threads x
// float4 = one coalesced 2KB row. This is the bandwidth-dominant kernel
// (~75 MB total traffic; embed gather hits L2 since the table fits in 192MB).
// ---------------------------------------------------------------------------
__global__ __launch_bounds__(128) void gather_concat_kernel(
    const float* __restrict__ embed, const int* __restrict__ src_tokens,
    const int* __restrict__ know_tokens, const int* __restrict__ src_mask,
    const int* __restrict__ know_mask, const int* __restrict__ sel_ids,
    float* __restrict__ out_enc, float* __restrict__ out_mask) {
  const int row = blockIdx.x;                // 0 .. N*TF-1
  const int n = row / TF;
  const int t = row - n * TF;
  int tok, mk;
  if (t < TK) {
    const int sel = sel_ids[n];
    const size_t base = ((size_t)n * K + sel) * TK + t;
    tok = know_tokens[base];
    mk  = know_mask[base];
  } else {
    const size_t base = (size_t)n * TS + (t - TK);
    tok = src_tokens[base];
    mk  = src_mask[base];
  }
  const int d0 = threadIdx.x * 4;
  float4 val = make_float4(0.f, 0.f, 0.f, 0.f);
  if (mk) val = *(const float4*)(embed + (size_t)tok * D + d0);
  *(float4*)(out_enc + (size_t)row * D + d0) = val;
  if (threadIdx.x == 0) out_mask[row] = mk ? 1.0f : 0.0f;
}

// ---------------------------------------------------------------------------
extern "C" void kernel_launch(void* const* d_in, const int* in_sizes, int n_in,
                              void* d_out, int out_size, void* d_ws, size_t ws_size,
                              hipStream_t stream) {
  (void)in_sizes; (void)n_in; (void)out_size; (void)ws_size;

  const float* embed       = (const float*)d_in[0];   // [V,D] f32
  const int*   src_tokens  = (const int*)d_in[1];     // [N,TS]
  const int*   know_tokens = (const int*)d_in[2];     // [N,K,TK]
  const int*   src_mask    = (const int*)d_in[3];     // [N,TS] bool->int
  const int*   know_mask   = (const int*)d_in[4];     // [N,K,TK] bool->int
  // d_in[5] ck_mask: unused by reference
  const int*   cs_ids      = (const int*)d_in[6];     // [N]
  const int*   use_cs_ids  = (const int*)d_in[7];     // scalar

  // Workspace layout (floats): ctx_use [N*D] | kn_use [N*K*D] | sel_ids [N]
  float* ws       = (float*)d_ws;
  float* ctx_use  = ws;
  float* kn_use   = ws + (size_t)N * D;
  int*   sel_ids  = (int*)(ws + (size_t)N * D + (size_t)N * K * D);

  // Output layout (floats): full_enc [N*TF*D] | full_mask [N*TF] | ck_attn [N*K]
  float* out_enc  = (float*)d_out;
  float* out_mask = out_enc + (size_t)N * TF * D;
  float* out_attn = out_mask + (size_t)N * TF;

  ctx_pool_kernel<<<dim3(N, 4), 128, 0, stream>>>(embed, src_tokens, src_mask, ctx_use);
  kn_pool_kernel<<<N * K, 128, 0, stream>>>(embed, know_tokens, know_mask, kn_use);
  ck_attn_wmma_kernel<<<N, 32, 0, stream>>>(kn_use, ctx_use, out_attn);
  select_id_kernel<<<N, 32, 0, stream>>>(out_attn, cs_ids, use_cs_ids, sel_ids);
  gather_concat_kernel<<<N * TF, 128, 0, stream>>>(embed, src_tokens, know_tokens,
                                                   src_mask, know_mask, sel_ids,
                                                   out_enc, out_mask);
}